// GCN_50096498540828
// MI455X (gfx1250) — compile-verified
//
#include <hip/hip_runtime.h>
#include <math.h>

#define N_NODES 80000
#define N_EDGES 1280000
#define IN_C 128
#define HID_C 64
#define OUT_C 40

typedef __attribute__((ext_vector_type(2))) float v2f;
typedef __attribute__((ext_vector_type(8))) float v8f;

// ---------------------------------------------------------------------------
// GEMM1: H1[80000,64] = X[80000,128] @ W1[128,64]    (f32 WMMA 16x16x4)
// One wave per 16-row tile; 4 N-tiles of 16 cols; K in steps of 4.
// A layout (16x4 f32): lane l -> row m = l%16, holds K = (l/16)*2 + {0,1}
// B layout (4x16 f32): lane l -> col n = l%16, holds K = (l/16)*2 + {0,1}
// C/D layout (16x16 f32): vgpr r, lane l -> M = r + 8*(l/16), N = l%16
// ---------------------------------------------------------------------------
__global__ __launch_bounds__(256) void gemm1_wmma(const float* __restrict__ X,
                                                  const float* __restrict__ W1,
                                                  float* __restrict__ H1) {
  const int wave = (blockIdx.x * blockDim.x + threadIdx.x) >> 5;  // 0..4999
  const int lane = threadIdx.x & 31;
  const int row0 = wave * 16;
  const int m = lane & 15;
  const int khalf = (lane >> 4) << 1;  // 0 or 2

  v8f acc0 = {}, acc1 = {}, acc2 = {}, acc3 = {};
  const float* xrow = X + (size_t)(row0 + m) * IN_C;

  for (int k0 = 0; k0 < IN_C; k0 += 4) {
    const int kb = k0 + khalf;
    v2f a = *(const v2f*)(xrow + kb);  // 8B-aligned (kb even)
    const float* wk0 = W1 + (size_t)kb * HID_C + m;  // K row kb,   col n=m
    const float* wk1 = wk0 + HID_C;                  // K row kb+1
    v2f b0 = {wk0[0],  wk1[0]};
    v2f b1 = {wk0[16], wk1[16]};
    v2f b2 = {wk0[32], wk1[32]};
    v2f b3 = {wk0[48], wk1[48]};
    acc0 = __builtin_amdgcn_wmma_f32_16x16x4_f32(false, a, false, b0, (short)0, acc0, false, false);
    acc1 = __builtin_amdgcn_wmma_f32_16x16x4_f32(false, a, false, b1, (short)0, acc1, false, false);
    acc2 = __builtin_amdgcn_wmma_f32_16x16x4_f32(false, a, false, b2, (short)0, acc2, false, false);
    acc3 = __builtin_amdgcn_wmma_f32_16x16x4_f32(false, a, false, b3, (short)0, acc3, false, false);
  }

  float* out = H1 + (size_t)(row0 + ((lane >> 4) << 3)) * HID_C + (lane & 15);
#pragma unroll
  for (int r = 0; r < 8; ++r) {
    out[(size_t)r * HID_C + 0]  = acc0[r];
    out[(size_t)r * HID_C + 16] = acc1[r];
    out[(size_t)r * HID_C + 32] = acc2[r];
    out[(size_t)r * HID_C + 48] = acc3[r];
  }
}

// ---------------------------------------------------------------------------
// GEMM2: H2[80000,40] = relu(A1[80000,64]) @ W2[64,40]   (3 N-tiles, masked)
// ---------------------------------------------------------------------------
__global__ __launch_bounds__(256) void gemm2_wmma(const float* __restrict__ A1,
                                                  const float* __restrict__ W2,
                                                  float* __restrict__ H2) {
  const int wave = (blockIdx.x * blockDim.x + threadIdx.x) >> 5;  // 0..4999
  const int lane = threadIdx.x & 31;
  const int row0 = wave * 16;
  const int m = lane & 15;
  const int khalf = (lane >> 4) << 1;

  v8f acc0 = {}, acc1 = {}, acc2 = {};
  const float* arow = A1 + (size_t)(row0 + m) * HID_C;
  const bool tail_ok = m < 8;  // cols 32..39 valid, 40..47 are padding

  for (int k0 = 0; k0 < HID_C; k0 += 4) {
    const int kb = k0 + khalf;
    v2f a = {fmaxf(arow[kb], 0.0f), fmaxf(arow[kb + 1], 0.0f)};  // fused ReLU
    const float* wk0 = W2 + (size_t)kb * OUT_C + m;
    const float* wk1 = wk0 + OUT_C;
    v2f b0 = {wk0[0], wk1[0]};
    v2f b1 = {wk0[16], wk1[16]};
    v2f b2 = {tail_ok ? wk0[32] : 0.0f, tail_ok ? wk1[32] : 0.0f};
    acc0 = __builtin_amdgcn_wmma_f32_16x16x4_f32(false, a, false, b0, (short)0, acc0, false, false);
    acc1 = __builtin_amdgcn_wmma_f32_16x16x4_f32(false, a, false, b1, (short)0, acc1, false, false);
    acc2 = __builtin_amdgcn_wmma_f32_16x16x4_f32(false, a, false, b2, (short)0, acc2, false, false);
  }

  const int n = lane & 15;
  float* out = H2 + (size_t)(row0 + ((lane >> 4) << 3)) * OUT_C + n;
#pragma unroll
  for (int r = 0; r < 8; ++r) {
    out[(size_t)r * OUT_C + 0]  = acc0[r];
    out[(size_t)r * OUT_C + 16] = acc1[r];
    if (n < 8) out[(size_t)r * OUT_C + 32] = acc2[r];
  }
}

// ---------------------------------------------------------------------------
// Edge-parallel scatter-add SpMM (channel-coalesced: lane == channel)
// ---------------------------------------------------------------------------
__global__ __launch_bounds__(256) void zero_f32(float* __restrict__ p, int n) {
  int i = blockIdx.x * blockDim.x + threadIdx.x;
  if (i < n) p[i] = 0.0f;
}

__global__ __launch_bounds__(256) void spmm_add_64(const int* __restrict__ ei,
                                                   const float* __restrict__ H,
                                                   float* __restrict__ A) {
  int i = blockIdx.x * blockDim.x + threadIdx.x;  // < N_EDGES*64
  int e = i >> 6;
  int c = i & 63;
  int src = ei[e];
  int dst = ei[N_EDGES + e];
  atomicAdd(&A[(size_t)dst * HID_C + c], H[(size_t)src * HID_C + c]);
}

__global__ __launch_bounds__(256) void spmm_add_40(const int* __restrict__ ei,
                                                   const float* __restrict__ H,
                                                   float* __restrict__ A) {
  int i = blockIdx.x * blockDim.x + threadIdx.x;  // < N_EDGES*40
  int e = i / OUT_C;
  int c = i - e * OUT_C;
  int src = ei[e];
  int dst = ei[N_EDGES + e];
  atomicAdd(&A[(size_t)dst * OUT_C + c], H[(size_t)src * OUT_C + c]);
}

// ---------------------------------------------------------------------------
// In-place log_softmax over 40 classes; one wave per node.
// lane handles channel `lane`, plus `lane+32` when lane < 8.
// ---------------------------------------------------------------------------
__global__ __launch_bounds__(256) void log_softmax_40(float* __restrict__ d) {
  const int wave = (blockIdx.x * blockDim.x + threadIdx.x) >> 5;  // node id
  const int lane = threadIdx.x & 31;
  float* base = d + (size_t)wave * OUT_C;

  float v0 = base[lane];
  float v1 = (lane < 8) ? base[32 + lane] : 0.0f;

  float mx = (lane < 8) ? fmaxf(v0, v1) : v0;
#pragma unroll
  for (int o = 16; o > 0; o >>= 1) mx = fmaxf(mx, __shfl_xor(mx, o, 32));

  float s = expf(v0 - mx) + ((lane < 8) ? expf(v1 - mx) : 0.0f);
#pragma unroll
  for (int o = 16; o > 0; o >>= 1) s += __shfl_xor(s, o, 32);

  float lse = mx + logf(s);
  base[lane] = v0 - lse;
  if (lane < 8) base[32 + lane] = v1 - lse;
}

// ---------------------------------------------------------------------------
extern "C" void kernel_launch(void* const* d_in, const int* in_sizes, int n_in,
                              void* d_out, int out_size, void* d_ws, size_t ws_size,
                              hipStream_t stream) {
  const float* X  = (const float*)d_in[0];
  const int*   EI = (const int*)d_in[1];
  const float* W1 = (const float*)d_in[2];
  const float* W2 = (const float*)d_in[3];
  float* out = (float*)d_out;

  float* H1 = (float*)d_ws;                       // 80000*64
  float* A1 = H1 + (size_t)N_NODES * HID_C;       // 80000*64
  float* H2 = A1 + (size_t)N_NODES * HID_C;       // 80000*40

  const int waves_total = N_NODES / 16;           // 5000
  const int gemm_blocks = waves_total / 8;        // 625 (exact)

  // Layer 1
  gemm1_wmma<<<gemm_blocks, 256, 0, stream>>>(X, W1, H1);
  {
    int n = N_NODES * HID_C;
    zero_f32<<<(n + 255) / 256, 256, 0, stream>>>(A1, n);
  }
  spmm_add_64<<<(N_EDGES * 64) / 256, 256, 0, stream>>>(EI, H1, A1);

  // Layer 2 (ReLU fused into GEMM2 A-load)
  gemm2_wmma<<<gemm_blocks, 256, 0, stream>>>(A1, W2, H2);
  {
    int n = N_NODES * OUT_C;
    zero_f32<<<(n + 255) / 256, 256, 0, stream>>>(out, n);
  }
  spmm_add_40<<<(N_EDGES * OUT_C) / 256, 256, 0, stream>>>(EI, H2, out);

  // log_softmax in place on d_out
  log_softmax_40<<<(N_NODES * 32) / 256, 256, 0, stream>>>(out);
}